// GaussianRBFNetworkR_67070209295133
// MI455X (gfx1250) — compile-verified
//
#include <hip/hip_runtime.h>

// CDNA5 / gfx1250 Gaussian RBF network, fully fp32 via V_WMMA_F32_16X16X4_F32.
//
// Factorization: P = L L^T  =>  U = X@L, V = centers@L,
//   q_x = rowsum(U^2), q_c = rowsum(V^2), cross = U @ V^T
//   out = sum_c w_c * exp(cross - 0.5 q_x - 0.5 q_c)
//
// 2x2 register blocking: each wave computes a 32x32 output block (4 WMMA
// accumulators) so each K=4 WMMA step costs 1 operand load instead of 2.
//
// ws layout (floats):
//   [0,      16384) : LT  (128x128)  LT[j][k] = L[k][j]
//   [16384,  81920) : V   (512x128)  = centers @ L
//   [81920,  82432) : tc  (512)      = -0.5 * q_c

typedef __attribute__((ext_vector_type(2))) float v2f;
typedef __attribute__((ext_vector_type(8))) float v8f;

#define F_DIM     128
#define N_SAMPLES 8192
#define C_CENTERS 512
#define M_TILE    64
#define U_STRIDE  132   // padded LDS stride (even -> 8B-aligned float2 loads)

__device__ __forceinline__ v8f wmma_f32_k4(v2f a, v2f b, v8f c) {
    // 8 args: (neg_a, A, neg_b, B, c_mod, C, reuse_a, reuse_b)
    return __builtin_amdgcn_wmma_f32_16x16x4_f32(false, a, false, b,
                                                 (short)0, c, false, false);
}

// ---------------------------------------------------------------- kernel 1 --
// LT[j][k] = L[k][j] = packed[k*(k+1)/2 + j] for k >= j else 0
__global__ void build_LT(const float* __restrict__ pe, float* __restrict__ LT) {
    int idx = blockIdx.x * blockDim.x + threadIdx.x;   // 0..16383
    int j = idx >> 7;
    int k = idx & 127;
    float v = 0.0f;
    if (k >= j) v = pe[(k * (k + 1)) / 2 + j];
    LT[idx] = v;
}

// ---------------------------------------------------------------- kernel 2 --
// V = centers @ L.  Block = 8 waves; wave w owns column tile w; block owns one
// 16-row band of centers. One 16x16 f32 tile per wave: 32 x WMMA_16x16x4.
__global__ void centers_V(const float* __restrict__ centers,
                          const float* __restrict__ LT,
                          float* __restrict__ V) {
    const int lane = threadIdx.x & 31;
    const int wave = threadIdx.x >> 5;      // 0..7
    const int row0 = blockIdx.x * 16;       // 32 blocks -> 512 rows
    const int col0 = wave * 16;
    const int m  = lane & 15;
    const int hl = lane >> 4;
    const int kh = hl << 1;

    const float* arow = centers + (size_t)(row0 + m) * F_DIM + kh;
    const float* brow = LT      + (size_t)(col0 + m) * F_DIM + kh;

    v8f acc = {};
#pragma unroll
    for (int kk = 0; kk < F_DIM; kk += 4) {
        v2f a = *(const v2f*)(arow + kk);
        v2f b = *(const v2f*)(brow + kk);
        acc = wmma_f32_k4(a, b, acc);
    }
#pragma unroll
    for (int v = 0; v < 8; ++v) {
        int r = row0 + v + 8 * hl;
        V[(size_t)r * F_DIM + col0 + m] = acc[v];
    }
}

// ---------------------------------------------------------------- kernel 3 --
// tc[c] = -0.5 * ||V[c]||^2
__global__ void qc_kernel(const float* __restrict__ V, float* __restrict__ tc) {
    int c = blockIdx.x * blockDim.x + threadIdx.x;
    if (c >= C_CENTERS) return;
    const float* row = V + (size_t)c * F_DIM;
    float s = 0.0f;
#pragma unroll 8
    for (int k = 0; k < F_DIM; ++k) { float x = row[k]; s = fmaf(x, x, s); }
    tc[c] = -0.5f * s;
}

// ---------------------------------------------------------------- kernel 4 --
// Fused main: per block, 64 rows of X.
//   Phase 1: U = Xtile @ L       (WMMA 2x2-blocked, U -> LDS)
//   Phase 2: qxh = -0.5 rowsum(U^2)
//   Phase 3: cross = U @ V^T     (WMMA 2x2-blocked from LDS), exp, reduce
__global__ void __launch_bounds__(256, 2)
rbf_main(const float* __restrict__ X,  const float* __restrict__ LT,
         const float* __restrict__ V,  const float* __restrict__ tc,
         const float* __restrict__ W,  float* __restrict__ out) {
    __shared__ float Ush[M_TILE * U_STRIDE];
    __shared__ float qxh[M_TILE];
    __shared__ float accum[M_TILE];

    const int tid  = threadIdx.x;
    const int lane = tid & 31;
    const int wave = tid >> 5;              // 0..7
    const int row0 = blockIdx.x * M_TILE;
    const int m  = lane & 15;
    const int hl = lane >> 4;
    const int kh = hl << 1;

    if (tid < M_TILE) accum[tid] = 0.0f;

    // ---- Phase 1: U tile (64x128) as 8 32x32 blocks, one per wave.
    //      wave: R = wave&1 -> row tiles {2R, 2R+1}; CP = wave>>1 -> cols.
    {
        const int rt0  = (wave & 1) * 32;        // row offset within tile
        const int col0 = (wave >> 1) * 32;       // column offset (0,32,64,96)
        const float* arow0 = X  + (size_t)(row0 + rt0 + m) * F_DIM + kh;
        const float* arow1 = arow0 + 16 * F_DIM;
        const float* brow0 = LT + (size_t)(col0 + m) * F_DIM + kh;
        const float* brow1 = brow0 + 16 * F_DIM;

        v8f acc00 = {}, acc01 = {}, acc10 = {}, acc11 = {};
#pragma unroll
        for (int kk = 0; kk < F_DIM; kk += 4) {
            v2f a0 = *(const v2f*)(arow0 + kk);
            v2f a1 = *(const v2f*)(arow1 + kk);
            v2f b0 = *(const v2f*)(brow0 + kk);
            v2f b1 = *(const v2f*)(brow1 + kk);
            acc00 = wmma_f32_k4(a0, b0, acc00);
            acc01 = wmma_f32_k4(a0, b1, acc01);
            acc10 = wmma_f32_k4(a1, b0, acc10);
            acc11 = wmma_f32_k4(a1, b1, acc11);
        }
#pragma unroll
        for (int v = 0; v < 8; ++v) {
            int r0 = rt0 + v + 8 * hl;
            Ush[(r0)      * U_STRIDE + col0 + m]      = acc00[v];
            Ush[(r0)      * U_STRIDE + col0 + 16 + m] = acc01[v];
            Ush[(r0 + 16) * U_STRIDE + col0 + m]      = acc10[v];
            Ush[(r0 + 16) * U_STRIDE + col0 + 16 + m] = acc11[v];
        }
    }
    __syncthreads();

    // ---- Phase 2: qxh[r] = -0.5 * sum_k U[r][k]^2
    for (int r = tid; r < M_TILE; r += 256) {
        const float* row = Ush + r * U_STRIDE;
        float s = 0.0f;
#pragma unroll 8
        for (int k = 0; k < F_DIM; ++k) { float x = row[k]; s = fmaf(x, x, s); }
        qxh[r] = -0.5f * s;
    }
    __syncthreads();

    // ---- Phase 3: cross (64x512) as 32 32x32 blocks; wave w handles
    //      wk = w, w+8, w+16, w+24  (R = wk&1, col-pair = wk>>1).
    for (int wk = wave; wk < 32; wk += 8) {
        const int rt0 = (wk & 1) * 32;           // row offset within tile
        const int c0  = (wk >> 1) * 32;          // center column base
        const float* urow0 = Ush + (rt0 + m) * U_STRIDE + kh;
        const float* urow1 = urow0 + 16 * U_STRIDE;
        const float* vrow0 = V + (size_t)(c0 + m) * F_DIM + kh;
        const float* vrow1 = vrow0 + 16 * F_DIM;

        const float tc0 = tc[c0 + m];
        const float tc1 = tc[c0 + 16 + m];
        const float wc0 = W[c0 + m];
        const float wc1 = W[c0 + 16 + m];

        v8f acc00 = {}, acc01 = {}, acc10 = {}, acc11 = {};
#pragma unroll
        for (int kk = 0; kk < F_DIM; kk += 4) {
            v2f a0 = *(const v2f*)(urow0 + kk);   // ds_load_b64 (conflict-free)
            v2f a1 = *(const v2f*)(urow1 + kk);
            v2f b0 = *(const v2f*)(vrow0 + kk);   // global_load_b64
            v2f b1 = *(const v2f*)(vrow1 + kk);
            acc00 = wmma_f32_k4(a0, b0, acc00);
            acc01 = wmma_f32_k4(a0, b1, acc01);
            acc10 = wmma_f32_k4(a1, b0, acc10);
            acc11 = wmma_f32_k4(a1, b1, acc11);
        }

#pragma unroll
        for (int v = 0; v < 8; ++v) {
            int r0 = rt0 + v + 8 * hl;
            float q0 = qxh[r0];
            float q1 = qxh[r0 + 16];
            float s0 = wc0 * __expf(acc00[v] + q0 + tc0)
                     + wc1 * __expf(acc01[v] + q0 + tc1);   // row r0 contribs
            float s1 = wc0 * __expf(acc10[v] + q1 + tc0)
                     + wc1 * __expf(acc11[v] + q1 + tc1);   // row r0+16 contribs
            // sum over the 16 center-columns held by this half-wave
            s0 += __shfl_xor(s0, 1, 32);
            s0 += __shfl_xor(s0, 2, 32);
            s0 += __shfl_xor(s0, 4, 32);
            s0 += __shfl_xor(s0, 8, 32);
            s1 += __shfl_xor(s1, 1, 32);
            s1 += __shfl_xor(s1, 2, 32);
            s1 += __shfl_xor(s1, 4, 32);
            s1 += __shfl_xor(s1, 8, 32);
            if (m == 0) {
                atomicAdd(&accum[r0],      s0);   // ds_add_f32
                atomicAdd(&accum[r0 + 16], s1);
            }
        }
    }
    __syncthreads();
    if (tid < M_TILE) out[row0 + tid] = accum[tid];
}

// ------------------------------------------------------------------ launch --
extern "C" void kernel_launch(void* const* d_in, const int* in_sizes, int n_in,
                              void* d_out, int out_size, void* d_ws, size_t ws_size,
                              hipStream_t stream) {
    const float* X       = (const float*)d_in[0];   // 8192*128
    const float* pe      = (const float*)d_in[1];   // 8256
    const float* centers = (const float*)d_in[2];   // 512*128
    const float* W       = (const float*)d_in[3];   // 512
    float* out = (float*)d_out;                     // 8192

    float* ws = (float*)d_ws;
    float* LT = ws;                     // 16384 floats
    float* V  = ws + 16384;             // 65536 floats
    float* tc = ws + 16384 + 65536;     // 512 floats

    build_LT <<<(F_DIM * F_DIM) / 256, 256, 0, stream>>>(pe, LT);
    centers_V<<<C_CENTERS / 16,        256, 0, stream>>>(centers, LT, V);
    qc_kernel<<<(C_CENTERS + 255)/256, 256, 0, stream>>>(V, tc);
    rbf_main <<<N_SAMPLES / M_TILE,    256, 0, stream>>>(X, LT, V, tc, W, out);
}